// GNNBackbone_81578608820912
// MI455X (gfx1250) — compile-verified
//
#include <hip/hip_runtime.h>

// Problem constants (fixed by the reference)
#define NB     8
#define NNODE  10000
#define NEDGE  160000
#define NN     (NB * NNODE)        // 80000 total nodes
#define TOTE   (NB * NEDGE)        // 1,280,000 total edges
#define HID    128
#define EMB    256
#define FIN    5
#define A_LAST 10
#define PROJ   (5 * HID + A_LAST + 1)   // 651

typedef __attribute__((ext_vector_type(2))) float v2f;
typedef __attribute__((ext_vector_type(8))) float v8f;

// ---------------------------------------------------------------------------
// Degree / normalization
// ---------------------------------------------------------------------------
__global__ void init_deg_kernel(float* __restrict__ deg) {
    int n = blockIdx.x * blockDim.x + threadIdx.x;
    if (n < NN) deg[n] = 1.0f;   // self-loop contributes 1 to every node
}

__global__ void count_deg_kernel(const int* __restrict__ edges,
                                 float* __restrict__ deg) {
    int e = blockIdx.x * blockDim.x + threadIdx.x;
    if (e >= TOTE) return;
    int b  = e / NEDGE;
    int ee = e - b * NEDGE;
    int dst = edges[b * 2 * NEDGE + NEDGE + ee] + b * NNODE;
    unsafeAtomicAdd(&deg[dst], 1.0f);
}

__global__ void dinv_kernel(float* __restrict__ deg) {
    int n = blockIdx.x * blockDim.x + threadIdx.x;
    if (n < NN) {
        float d = deg[n];
        deg[n] = (d > 0.0f) ? __frsqrt_rn(d) : 0.0f;
    }
}

// ---------------------------------------------------------------------------
// Layer-1 GEMM: h = x @ W1^T   (K = 5, scalar is the right tool)
// ---------------------------------------------------------------------------
__global__ void gemm1_kernel(const float* __restrict__ X,
                             const float* __restrict__ W1,
                             float* __restrict__ H) {
    int idx = blockIdx.x * blockDim.x + threadIdx.x;  // over NN*HID
    if (idx >= NN * HID) return;
    int n = idx >> 7;
    int j = idx & 127;
    const float* xr = X  + (size_t)n * FIN;
    const float* wr = W1 + (size_t)j * FIN;
    float acc = 0.0f;
#pragma unroll
    for (int k = 0; k < FIN; ++k) acc = fmaf(xr[k], wr[k], acc);
    H[idx] = acc;
}

// ---------------------------------------------------------------------------
// Self-loop term: out[n,:] = dinv[n]^2 * h[n,:]   (also inits accumulator)
// ---------------------------------------------------------------------------
__global__ void selfloop_kernel(const float* __restrict__ H,
                                const float* __restrict__ dinv,
                                float* __restrict__ OUT) {
    int idx = blockIdx.x * blockDim.x + threadIdx.x;
    if (idx >= NN * HID) return;
    int n = idx >> 7;
    float di = dinv[n];
    OUT[idx] = H[idx] * di * di;
}

// ---------------------------------------------------------------------------
// Edge scatter: one wave32 per edge.
// Lane l: float4 gather of h[src, 4l..4l+3], 4 native f32 L2 atomics to dst.
// ---------------------------------------------------------------------------
__global__ void __launch_bounds__(256)
edge_scatter_kernel(const int* __restrict__ edges,
                    const float* __restrict__ dinv,
                    const float* __restrict__ H,
                    float* __restrict__ OUT) {
    int tid  = threadIdx.x;
    int lane = tid & 31;
    int wave = tid >> 5;
    int e = blockIdx.x * 8 + wave;          // 8 waves/block, 1 edge/wave
    if (e >= TOTE) return;
    int b  = e / NEDGE;
    int ee = e - b * NEDGE;
    const int* eb = edges + (size_t)b * 2 * NEDGE;
    int src = eb[ee] + b * NNODE;
    int dst = eb[NEDGE + ee] + b * NNODE;
    float norm = dinv[src] * dinv[dst];

    const float* hs = H   + (size_t)src * HID + lane * 4;
    float*       od = OUT + (size_t)dst * HID + lane * 4;
    __builtin_prefetch(od, 1, 0);           // global_prefetch of RMW target
    float4 v = *(const float4*)hs;
    unsafeAtomicAdd(od + 0, v.x * norm);
    unsafeAtomicAdd(od + 1, v.y * norm);
    unsafeAtomicAdd(od + 2, v.z * norm);
    unsafeAtomicAdd(od + 3, v.w * norm);
}

// ---------------------------------------------------------------------------
// Bias + ReLU (in place)
// ---------------------------------------------------------------------------
__global__ void bias_relu_kernel(float* __restrict__ X,
                                 const float* __restrict__ bias) {
    int idx = blockIdx.x * blockDim.x + threadIdx.x;
    if (idx >= NN * HID) return;
    float v = X[idx] + bias[idx & 127];
    X[idx] = v > 0.0f ? v : 0.0f;
}

// ---------------------------------------------------------------------------
// Layer-2 GEMM via f32 WMMA: H = X1 @ W2^T, [80000,128] x [128,128].
// Block = 256 threads (8 waves), block owns a 128-row strip (8 M-tiles).
// W2 staged transposed in 64KB LDS; each wave keeps its A strip in regs.
// ---------------------------------------------------------------------------
__global__ void __launch_bounds__(256)
gemm2_wmma_kernel(const float* __restrict__ X,
                  const float* __restrict__ W2,
                  float* __restrict__ H) {
    __shared__ float Bs[HID * HID];   // Bs[k*128 + j] = W2[j,k]  (64 KB)
    int tid = threadIdx.x;
    // Coalesced LDS writes; strided global reads absorbed by L2.
    for (int o = tid; o < HID * HID; o += 256) {
        int k = o >> 7, j = o & 127;
        Bs[o] = W2[j * HID + k];
    }
    __syncthreads();

    int lane = tid & 31;
    int wave = tid >> 5;
    int m0   = blockIdx.x * 128 + wave * 16;
    int mrow = m0 + (lane & 15);
    int kl   = (lane >> 4) << 1;            // lanes 0-15: K+{0,1}; 16-31: K+{2,3}

    // Hoist the 16x128 A strip for this wave: 32 K-steps of v2f per lane.
    const float* xrow = X + (size_t)mrow * HID;
    v2f afrag[32];
#pragma unroll
    for (int kk = 0; kk < 32; ++kk) {
        afrag[kk].x = xrow[kk * 4 + kl];
        afrag[kk].y = xrow[kk * 4 + kl + 1];
    }

    int mbase = m0 + ((lane >> 4) << 3);    // C/D: lanes 16-31 hold M=8..15
    for (int nt = 0; nt < 8; ++nt) {
        int col = nt * 16 + (lane & 15);
        v8f c = {0.f, 0.f, 0.f, 0.f, 0.f, 0.f, 0.f, 0.f};
#pragma unroll
        for (int kk = 0; kk < 32; ++kk) {
            v2f bf;
            bf.x = Bs[(kk * 4 + kl) * HID + col];
            bf.y = Bs[(kk * 4 + kl + 1) * HID + col];
            c = __builtin_amdgcn_wmma_f32_16x16x4_f32(
                    false, afrag[kk], false, bf, (short)0, c, false, false);
        }
#pragma unroll
        for (int r = 0; r < 8; ++r)
            H[(size_t)(mbase + r) * HID + col] = c[r];
    }
}

// ---------------------------------------------------------------------------
// Head: gather 5 node embeddings + actions + step, then 2 dense layers.
// One block per batch element.
// ---------------------------------------------------------------------------
__global__ void __launch_bounds__(256)
head_kernel(const float* __restrict__ X2,
            const int* __restrict__ position_seq,
            const int* __restrict__ available_moves,
            const float* __restrict__ actions,
            const float* __restrict__ steps,
            const float* __restrict__ Wp1, const float* __restrict__ bp1,
            const float* __restrict__ Wp2, const float* __restrict__ bp2,
            float* __restrict__ out) {
    __shared__ float cx[PROJ];
    __shared__ float hl[EMB];
    int b   = blockIdx.x;
    int tid = threadIdx.x;

    for (int i = tid; i < PROJ; i += 256) {
        float v;
        if (i < 5 * HID) {
            int slot = i >> 7;              // which of the 5 positions
            int j    = i & 127;
            int p = (slot < 4) ? available_moves[b * 4 + slot]
                               : position_seq[b];
            p += b * NNODE;
            v = X2[(size_t)p * HID + j];
        } else if (i < 5 * HID + A_LAST) {
            v = actions[b * A_LAST + (i - 5 * HID)];
        } else {
            v = steps[b];
        }
        cx[i] = v;
    }
    __syncthreads();

    // layer 1: [651] -> [256], relu
    {
        float acc = bp1[tid];
        const float* wr = Wp1 + (size_t)tid * PROJ;
        for (int k = 0; k < PROJ; ++k) acc = fmaf(cx[k], wr[k], acc);
        hl[tid] = acc > 0.0f ? acc : 0.0f;
    }
    __syncthreads();

    // layer 2: [256] -> [256]
    {
        float acc = bp2[tid];
        const float* wr = Wp2 + (size_t)tid * EMB;
#pragma unroll 4
        for (int k = 0; k < EMB; ++k) acc = fmaf(hl[k], wr[k], acc);
        out[b * EMB + tid] = acc;
    }
}

// ---------------------------------------------------------------------------
// Launcher
// ---------------------------------------------------------------------------
extern "C" void kernel_launch(void* const* d_in, const int* in_sizes, int n_in,
                              void* d_out, int out_size, void* d_ws, size_t ws_size,
                              hipStream_t stream) {
    const float* graph_x         = (const float*)d_in[0];
    const int*   graph_edges     = (const int*)  d_in[1];
    const int*   position_seq    = (const int*)  d_in[2];
    const int*   available_moves = (const int*)  d_in[3];
    const float* actions         = (const float*)d_in[4];
    const float* steps           = (const float*)d_in[5];
    const float* W1  = (const float*)d_in[6];
    const float* b1  = (const float*)d_in[7];
    const float* W2  = (const float*)d_in[8];
    const float* b2  = (const float*)d_in[9];
    const float* Wp1 = (const float*)d_in[10];
    const float* bp1 = (const float*)d_in[11];
    const float* Wp2 = (const float*)d_in[12];
    const float* bp2 = (const float*)d_in[13];
    float* out = (float*)d_out;

    char* ws = (char*)d_ws;
    float* dinv = (float*)ws;                                   // 320 KB
    float* bufA = (float*)(ws + ((size_t)4  << 20));            // 40.96 MB
    float* bufB = (float*)(ws + ((size_t)48 << 20));            // 40.96 MB

    const int T = 256;
    const int gNodes = (NN + T - 1) / T;             // 313
    const int gFeat  = (NN * HID + T - 1) / T;       // 40000
    const int gEdge  = (TOTE + T - 1) / T;           // 5000
    const int gScat  = TOTE / 8;                     // 160000 (wave per edge)
    const int gGemm2 = NN / 128;                     // 625

    // normalization
    init_deg_kernel<<<gNodes, T, 0, stream>>>(dinv);
    count_deg_kernel<<<gEdge, T, 0, stream>>>(graph_edges, dinv);
    dinv_kernel<<<gNodes, T, 0, stream>>>(dinv);

    // GCN layer 1
    gemm1_kernel<<<gFeat, T, 0, stream>>>(graph_x, W1, bufA);
    selfloop_kernel<<<gFeat, T, 0, stream>>>(bufA, dinv, bufB);
    edge_scatter_kernel<<<gScat, T, 0, stream>>>(graph_edges, dinv, bufA, bufB);
    bias_relu_kernel<<<gFeat, T, 0, stream>>>(bufB, b1);

    // GCN layer 2 (WMMA GEMM)
    gemm2_wmma_kernel<<<gGemm2, T, 0, stream>>>(bufB, W2, bufA);
    selfloop_kernel<<<gFeat, T, 0, stream>>>(bufA, dinv, bufB);
    edge_scatter_kernel<<<gScat, T, 0, stream>>>(graph_edges, dinv, bufA, bufB);
    bias_relu_kernel<<<gFeat, T, 0, stream>>>(bufB, b2);

    // head
    head_kernel<<<NB, T, 0, stream>>>(bufB, position_seq, available_moves,
                                      actions, steps, Wp1, bp1, Wp2, bp2, out);
}